// LIAALayerConfig_68917045231949
// MI455X (gfx1250) — compile-verified
//
#include <hip/hip_runtime.h>
#include <math.h>

#define EPSV   1e-12f
#define LNEPS  1e-5f
#define RIDGE  0.01f
#define MDIM   256
#define LDIM   4096
#define TDIM   2048

typedef __attribute__((ext_vector_type(2))) float v2f;
typedef __attribute__((ext_vector_type(8))) float v8f;
typedef unsigned int v4u __attribute__((ext_vector_type(4)));
typedef int v8ii __attribute__((ext_vector_type(8)));
typedef int v4ii __attribute__((ext_vector_type(4)));

__device__ __forceinline__ v8f wmma_f32(v2f a, v2f b, v8f c) {
  // D = A(16x4) * B(4x16) + C, fp32 tensor op on CDNA5
  return __builtin_amdgcn_wmma_f32_16x16x4_f32(false, a, false, b, (short)0, c,
                                               false, false);
}

// ---------------------------------------------------------------------------
// TDM: DMA a [tile_rows x tile_cols] f32 tile (row-major, row stride
// row_stride_elems) from global memory into LDS at byte offset lds_off.
// D# layout per CDNA5 ISA ch.8 (group0: count/lds/global/type; group1:
// data_size=4B, tensor dims, tile dims, dim0 stride). 2D -> groups 2/3 zero.
// This toolchain exposes the 6-arg builtin (extra int32x8 before cpol).
// Must be issued by a single wave; caller waits TENSORcnt then barriers.
// ---------------------------------------------------------------------------
__device__ __forceinline__ void tdm_load_f32_2d(const void* gptr, unsigned lds_off,
                                                unsigned tile_cols, unsigned tile_rows,
                                                unsigned row_stride_elems) {
  unsigned long long ga = (unsigned long long)(uintptr_t)gptr;
  v4u g0;
  g0.x = 1u;                                                   // count=1
  g0.y = lds_off;                                              // lds_addr
  g0.z = (unsigned)ga;                                         // global_addr lo
  g0.w = (unsigned)((ga >> 32) & 0x01FFFFFFu) | (2u << 30);    // addr hi | type=2
  v8ii g1;
  g1[0] = (2 << 16);                                           // data_size = 4B
  g1[1] = (int)((row_stride_elems & 0xFFFFu) << 16);           // tensor_dim0 lo16
  g1[2] = (int)((row_stride_elems >> 16) |
                ((tile_rows & 0xFFFFu) << 16));                // dim0 hi | dim1 lo
  g1[3] = (int)((tile_rows >> 16) | (tile_cols << 16));        // dim1 hi | tile_dim0
  g1[4] = (int)(tile_rows & 0xFFFFu);                          // tile_dim1 (tile_dim2=0)
  g1[5] = (int)row_stride_elems;                               // dim0_stride lo32
  g1[6] = 0;                                                   // dim0_stride hi | dim1_stride lo
  g1[7] = 0;
  v4ii gz4 = {0, 0, 0, 0};
  v8ii gz8 = {0, 0, 0, 0, 0, 0, 0, 0};
  __builtin_amdgcn_tensor_load_to_lds(g0, g1, gz4, gz4, gz8, 0);
}

// ---------------------------------------------------------------------------
// C[i,j] = sum_k A[i,k] * conj(B[j,k])  (+ ridge on diagonal)
// A,B complex [N x K] row-major (planar re/im). One wave per 16x32 tile:
// A fragments reused across two B sub-tiles -> 8 wmma per 6 b64 loads.
// conj(B):  Cre = rr + ii ; Cim = ir - ri
// ---------------------------------------------------------------------------
__global__ void gemm_nt_conj(const float* __restrict__ Ar, const float* __restrict__ Ai,
                             const float* __restrict__ Br, const float* __restrict__ Bi,
                             float* __restrict__ Cre, float* __restrict__ Cim,
                             int N, int K, float ridge) {
  const int lane = threadIdx.x & 31;
  const int half = lane >> 4;
  const int l16  = lane & 15;
  const int i0 = blockIdx.y * 16;
  const int j0 = blockIdx.x * 32;
  const int ka = 2 * half;
  v8f rr0 = {}, ii0 = {}, ir0 = {}, ri0 = {};
  v8f rr1 = {}, ii1 = {}, ir1 = {}, ri1 = {};
  const float* arow_r = Ar + (size_t)(i0 + l16) * K + ka;
  const float* arow_i = Ai + (size_t)(i0 + l16) * K + ka;
  const float* b0row_r = Br + (size_t)(j0 + l16) * K + ka;
  const float* b0row_i = Bi + (size_t)(j0 + l16) * K + ka;
  const float* b1row_r = Br + (size_t)(j0 + 16 + l16) * K + ka;
  const float* b1row_i = Bi + (size_t)(j0 + 16 + l16) * K + ka;
  for (int k0 = 0; k0 < K; k0 += 4) {
    v2f aR, aI, b0R, b0I, b1R, b1I;
    aR.x  = arow_r[k0];     aR.y  = arow_r[k0 + 1];
    aI.x  = arow_i[k0];     aI.y  = arow_i[k0 + 1];
    b0R.x = b0row_r[k0];    b0R.y = b0row_r[k0 + 1];
    b0I.x = b0row_i[k0];    b0I.y = b0row_i[k0 + 1];
    b1R.x = b1row_r[k0];    b1R.y = b1row_r[k0 + 1];
    b1I.x = b1row_i[k0];    b1I.y = b1row_i[k0 + 1];
    rr0 = wmma_f32(aR, b0R, rr0);
    ii0 = wmma_f32(aI, b0I, ii0);
    ir0 = wmma_f32(aI, b0R, ir0);
    ri0 = wmma_f32(aR, b0I, ri0);
    rr1 = wmma_f32(aR, b1R, rr1);
    ii1 = wmma_f32(aI, b1I, ii1);
    ir1 = wmma_f32(aI, b1R, ir1);
    ri1 = wmma_f32(aR, b1I, ri1);
  }
  for (int v = 0; v < 8; ++v) {
    int row  = i0 + v + 8 * half;
    int col0 = j0 + l16;
    int col1 = j0 + 16 + l16;
    float cre0 = rr0[v] + ii0[v];
    float cim0 = ir0[v] - ri0[v];
    float cre1 = rr1[v] + ii1[v];
    float cim1 = ir1[v] - ri1[v];
    if (row == col0) cre0 += ridge;
    if (row == col1) cre1 += ridge;
    Cre[(size_t)row * N + col0] = cre0;
    Cim[(size_t)row * N + col0] = cim0;
    Cre[(size_t)row * N + col1] = cre1;
    Cim[(size_t)row * N + col1] = cim1;
  }
}

// ---------------------------------------------------------------------------
// C[l,m] = sum_k conj(A[k,l]) * B[k,m]; A: [256 x 4096], B: [256 x 256]
// B 256x16 panel staged in LDS once per block via TDM; 8 waves / 8 row tiles.
// conj(A):  Cre = rr + ii ; Cim = ri - ir
// ---------------------------------------------------------------------------
__global__ void gemm_ah_b(const float* __restrict__ Ar, const float* __restrict__ Ai,
                          const float* __restrict__ Br, const float* __restrict__ Bi,
                          float* __restrict__ Cre, float* __restrict__ Cim) {
  __shared__ float Bre_s[MDIM * 16];
  __shared__ float Bim_s[MDIM * 16];
  const int tid  = threadIdx.x;
  const int lane = tid & 31;
  const int wave = tid >> 5;
  const int half = lane >> 4;
  const int l16  = lane & 15;
  const int m0 = blockIdx.x * 16;
  const int l0 = blockIdx.y * 128 + wave * 16;
  const int ka = 2 * half;
  if (tid < 32) {
    tdm_load_f32_2d(Br + m0, (unsigned)(size_t)(void*)Bre_s, 16, MDIM, MDIM);
    tdm_load_f32_2d(Bi + m0, (unsigned)(size_t)(void*)Bim_s, 16, MDIM, MDIM);
    __builtin_amdgcn_s_wait_tensorcnt(0);
  }
  __syncthreads();
  v8f rr = {}, ii = {}, ir = {}, ri = {};
  for (int k0 = 0; k0 < MDIM; k0 += 4) {
    v2f aR, aI, bR, bI;
    aR.x = Ar[(size_t)(k0 + ka)     * LDIM + l0 + l16];
    aR.y = Ar[(size_t)(k0 + ka + 1) * LDIM + l0 + l16];
    aI.x = Ai[(size_t)(k0 + ka)     * LDIM + l0 + l16];
    aI.y = Ai[(size_t)(k0 + ka + 1) * LDIM + l0 + l16];
    bR.x = Bre_s[(k0 + ka) * 16 + l16];
    bR.y = Bre_s[(k0 + ka + 1) * 16 + l16];
    bI.x = Bim_s[(k0 + ka) * 16 + l16];
    bI.y = Bim_s[(k0 + ka + 1) * 16 + l16];
    rr = wmma_f32(aR, bR, rr);
    ii = wmma_f32(aI, bI, ii);
    ir = wmma_f32(aI, bR, ir);
    ri = wmma_f32(aR, bI, ri);
  }
  for (int v = 0; v < 8; ++v) {
    int row = l0 + v + 8 * half;
    int col = m0 + l16;
    Cre[(size_t)row * MDIM + col] = rr[v] + ii[v];
    Cim[(size_t)row * MDIM + col] = ri[v] - ir[v];
  }
}

// ---------------------------------------------------------------------------
// C[l,m] = sum_k A[l,k] * B[k,m]; A: [4096 x 256], B: [256 x 256]
// Same TDM-staged B panel. plain complex: Cre = rr - ii ; Cim = ri + ir
// ---------------------------------------------------------------------------
__global__ void gemm_nn(const float* __restrict__ Ar, const float* __restrict__ Ai,
                        const float* __restrict__ Br, const float* __restrict__ Bi,
                        float* __restrict__ Cre, float* __restrict__ Cim) {
  __shared__ float Bre_s[MDIM * 16];
  __shared__ float Bim_s[MDIM * 16];
  const int tid  = threadIdx.x;
  const int lane = tid & 31;
  const int wave = tid >> 5;
  const int half = lane >> 4;
  const int l16  = lane & 15;
  const int m0 = blockIdx.x * 16;
  const int l0 = blockIdx.y * 128 + wave * 16;
  const int ka = 2 * half;
  if (tid < 32) {
    tdm_load_f32_2d(Br + m0, (unsigned)(size_t)(void*)Bre_s, 16, MDIM, MDIM);
    tdm_load_f32_2d(Bi + m0, (unsigned)(size_t)(void*)Bim_s, 16, MDIM, MDIM);
    __builtin_amdgcn_s_wait_tensorcnt(0);
  }
  __syncthreads();
  v8f rr = {}, ii = {}, ir = {}, ri = {};
  const float* arow_r = Ar + (size_t)(l0 + l16) * MDIM + ka;
  const float* arow_i = Ai + (size_t)(l0 + l16) * MDIM + ka;
  for (int k0 = 0; k0 < MDIM; k0 += 4) {
    v2f aR, aI, bR, bI;
    aR.x = arow_r[k0];  aR.y = arow_r[k0 + 1];
    aI.x = arow_i[k0];  aI.y = arow_i[k0 + 1];
    bR.x = Bre_s[(k0 + ka) * 16 + l16];
    bR.y = Bre_s[(k0 + ka + 1) * 16 + l16];
    bI.x = Bim_s[(k0 + ka) * 16 + l16];
    bI.y = Bim_s[(k0 + ka + 1) * 16 + l16];
    rr = wmma_f32(aR, bR, rr);
    ii = wmma_f32(aI, bI, ii);
    ir = wmma_f32(aI, bR, ir);
    ri = wmma_f32(aR, bI, ri);
  }
  for (int v = 0; v < 8; ++v) {
    int row = l0 + v + 8 * half;
    int col = m0 + l16;
    Cre[(size_t)row * MDIM + col] = rr[v] - ii[v];
    Cim[(size_t)row * MDIM + col] = ri[v] + ir[v];
  }
}

// AP = A * diag(p) (elementwise row scale), keeps the GEMM inner loop clean.
__global__ void scale_ap(const float* __restrict__ Ar, const float* __restrict__ Ai,
                         const float* __restrict__ p,
                         float* __restrict__ APr, float* __restrict__ APi) {
  size_t i = (size_t)blockIdx.x * blockDim.x + threadIdx.x;
  int l = (int)(i & (LDIM - 1));
  float s = p[l];
  APr[i] = Ar[i] * s;
  APi[i] = Ai[i] * s;
}

// ---------------------------------------------------------------------------
// Complex Gauss-Jordan inverse of a 256x256 matrix; 1 block of 256 threads,
// thread t owns row t; pivot row staged in LDS.
// ---------------------------------------------------------------------------
__global__ void invert_gj(const float* __restrict__ Rre, const float* __restrict__ Rim,
                          float* __restrict__ augre, float* __restrict__ augim,
                          float* __restrict__ RIre, float* __restrict__ RIim) {
  const int t = threadIdx.x;              // 0..255, owns row t
  __shared__ float pivre[2 * MDIM];
  __shared__ float pivim[2 * MDIM];
  for (int c = 0; c < MDIM; ++c) {
    augre[(size_t)t * 512 + c] = Rre[(size_t)t * MDIM + c];
    augim[(size_t)t * 512 + c] = Rim[(size_t)t * MDIM + c];
    augre[(size_t)t * 512 + MDIM + c] = (c == t) ? 1.0f : 0.0f;
    augim[(size_t)t * 512 + MDIM + c] = 0.0f;
  }
  __syncthreads();
  for (int col = 0; col < MDIM; ++col) {
    float pre = augre[(size_t)col * 512 + col];
    float pim = augim[(size_t)col * 512 + col];
    float inv = 1.0f / (pre * pre + pim * pim);
    float ire =  pre * inv;
    float iim = -pim * inv;                 // 1/p = conj(p)/|p|^2
    for (int c = t; c < 512; c += 256) {
      float xr = augre[(size_t)col * 512 + c];
      float xi = augim[(size_t)col * 512 + c];
      pivre[c] = xr * ire - xi * iim;
      pivim[c] = xr * iim + xi * ire;
    }
    __syncthreads();
    for (int c = t; c < 512; c += 256) {    // write back normalized pivot row
      augre[(size_t)col * 512 + c] = pivre[c];
      augim[(size_t)col * 512 + c] = pivim[c];
    }
    if (t != col) {                          // eliminate own row
      float fr = augre[(size_t)t * 512 + col];
      float fi = augim[(size_t)t * 512 + col];
      for (int c = 0; c < 512; ++c) {
        float pr = pivre[c], pi = pivim[c];
        augre[(size_t)t * 512 + c] -= fr * pr - fi * pi;
        augim[(size_t)t * 512 + c] -= fr * pi + fi * pr;
      }
    }
    __syncthreads();
  }
  for (int c = 0; c < MDIM; ++c) {
    RIre[(size_t)t * MDIM + c] = augre[(size_t)t * 512 + MDIM + c];
    RIim[(size_t)t * MDIM + c] = augim[(size_t)t * 512 + MDIM + c];
  }
}

// ---------------------------------------------------------------------------
// q[l] = sum_m Re( W[l,m] * conj(ARi[l,m]) )   (= rowwise ||S||^2 via Gram)
// d[l] = sum_m Re( ARi[l,m] * conj(A[m,l]) )
// ---------------------------------------------------------------------------
__global__ void qd_kernel(const float* __restrict__ Wre, const float* __restrict__ Wim,
                          const float* __restrict__ ARre, const float* __restrict__ ARim,
                          const float* __restrict__ Are, const float* __restrict__ Aim,
                          float* __restrict__ q, float* __restrict__ d) {
  const int l = blockIdx.x;
  const int t = threadIdx.x;               // 256 threads == MDIM
  float wr = Wre[(size_t)l * MDIM + t], wi = Wim[(size_t)l * MDIM + t];
  float ar = ARre[(size_t)l * MDIM + t], ai = ARim[(size_t)l * MDIM + t];
  float qp = wr * ar + wi * ai;
  float dp = ar * Are[(size_t)t * LDIM + l] + ai * Aim[(size_t)t * LDIM + l];
  __shared__ float sq[256], sd[256];
  sq[t] = qp; sd[t] = dp;
  __syncthreads();
  for (int s = 128; s > 0; s >>= 1) {
    if (t < s) { sq[t] += sq[t + s]; sd[t] += sd[t + s]; }
    __syncthreads();
  }
  if (t == 0) { q[l] = sq[0]; d[l] = sd[0]; }
}

__global__ void uv_kernel(const float* __restrict__ gamma, const float* __restrict__ q,
                          const float* __restrict__ d, const float* __restrict__ delta,
                          float* __restrict__ v, float* __restrict__ u) {
  int i = blockIdx.x * blockDim.x + threadIdx.x;
  if (i >= LDIM) return;
  float dl = 1.0f / (1.0f + __expf(-delta[0]));
  float vi = q[i] / (d[i] + EPSV);
  v[i] = vi;
  u[i] = gamma[i] + dl * (vi - gamma[i]);
}

// LayerNorm over full length-L vector, then q/k/v projections (scalar embed).
__global__ void ln_prep(const float* __restrict__ u, const float* __restrict__ ln_w,
                        const float* __restrict__ ln_b, const float* __restrict__ in_w,
                        const float* __restrict__ in_b,
                        float* __restrict__ qv, float* __restrict__ kv,
                        float* __restrict__ vv) {
  __shared__ float ss[1024], s2[1024];
  const int t = threadIdx.x;
  float a = 0.0f, b = 0.0f;
  for (int i = t; i < LDIM; i += 1024) { float x = u[i]; a += x; b += x * x; }
  ss[t] = a; s2[t] = b;
  __syncthreads();
  for (int s = 512; s > 0; s >>= 1) {
    if (t < s) { ss[t] += ss[t + s]; s2[t] += s2[t + s]; }
    __syncthreads();
  }
  float mu   = ss[0] / (float)LDIM;
  float var  = s2[0] / (float)LDIM - mu * mu;
  float rstd = rsqrtf(var + LNEPS);
  float w0 = in_w[0], w1 = in_w[1], w2 = in_w[2];
  float b0 = in_b[0], b1 = in_b[1], b2 = in_b[2];
  for (int i = t; i < LDIM; i += 1024) {
    float up = (u[i] - mu) * rstd * ln_w[i] + ln_b[i];
    qv[i] = up * w0 + b0;
    kv[i] = up * w1 + b1;
    vv[i] = up * w2 + b2;
  }
}

// Rank-1 score attention: attn_out[i] = softmax_j(qv_i*kv_j) . vv * out_w + out_b
__global__ void attn_kernel(const float* __restrict__ qv, const float* __restrict__ kv,
                            const float* __restrict__ vv, const float* __restrict__ out_w,
                            const float* __restrict__ out_b, float* __restrict__ attn) {
  const int i = blockIdx.x;
  const int t = threadIdx.x;              // 256
  float qi = qv[i];
  __shared__ float sm[256];
  float mx = -3.4028235e38f;
  for (int j = t; j < LDIM; j += 256) mx = fmaxf(mx, qi * kv[j]);
  sm[t] = mx;
  __syncthreads();
  for (int s = 128; s > 0; s >>= 1) {
    if (t < s) sm[t] = fmaxf(sm[t], sm[t + s]);
    __syncthreads();
  }
  float m = sm[0];
  __syncthreads();
  __shared__ float sz[256], sn[256];
  float z = 0.0f, n = 0.0f;
  for (int j = t; j < LDIM; j += 256) {
    float e = __expf(qi * kv[j] - m);
    z += e;
    n += e * vv[j];
  }
  sz[t] = z; sn[t] = n;
  __syncthreads();
  for (int s = 128; s > 0; s >>= 1) {
    if (t < s) { sz[t] += sz[t + s]; sn[t] += sn[t + s]; }
    __syncthreads();
  }
  if (t == 0) attn[i] = (sn[0] / sz[0]) * out_w[0] + out_b[0];
}

// g = sigmoid(gate_W @ u + gate_b); r = g*v + (1-g)*p; out = relu(r + attn - lmbda)
__global__ void gate_final(const float* __restrict__ gW, const float* __restrict__ gb,
                           const float* __restrict__ u, const float* __restrict__ v,
                           const float* __restrict__ gamma, const float* __restrict__ attn,
                           const float* __restrict__ lmbda, float* __restrict__ out) {
  const int row = blockIdx.x;
  const int t = threadIdx.x;              // 256
  const float* wrow = gW + (size_t)row * LDIM;
  float acc = 0.0f;
  for (int c = t; c < LDIM; c += 256) acc += wrow[c] * u[c];
  __shared__ float s[256];
  s[t] = acc;
  __syncthreads();
  for (int st = 128; st > 0; st >>= 1) {
    if (t < st) s[t] += s[t + st];
    __syncthreads();
  }
  if (t == 0) {
    float g = 1.0f / (1.0f + __expf(-(s[0] + gb[row])));
    float r = g * v[row] + (1.0f - g) * gamma[row];
    float sr = r + attn[row];
    out[row] = fmaxf(sr - lmbda[0], 0.0f);
  }
}

extern "C" void kernel_launch(void* const* d_in, const int* in_sizes, int n_in,
                              void* d_out, int out_size, void* d_ws, size_t ws_size,
                              hipStream_t stream) {
  (void)in_sizes; (void)n_in; (void)out_size; (void)ws_size;
  const float* gamma  = (const float*)d_in[0];
  const float* A_re   = (const float*)d_in[1];
  const float* A_im   = (const float*)d_in[2];
  const float* X_re   = (const float*)d_in[3];
  const float* X_im   = (const float*)d_in[4];
  const float* ln_w   = (const float*)d_in[5];
  const float* ln_b   = (const float*)d_in[6];
  const float* in_w   = (const float*)d_in[7];
  const float* in_b   = (const float*)d_in[8];
  const float* out_w  = (const float*)d_in[9];
  const float* out_b  = (const float*)d_in[10];
  const float* gate_W = (const float*)d_in[11];
  const float* gate_b = (const float*)d_in[12];
  const float* delta  = (const float*)d_in[13];
  const float* lmbda  = (const float*)d_in[14];
  float* out = (float*)d_out;

  float* ws = (float*)d_ws;
  size_t o = 0;
  float* APre  = ws + o; o += (size_t)MDIM * LDIM;
  float* APim  = ws + o; o += (size_t)MDIM * LDIM;
  float* Rre   = ws + o; o += (size_t)MDIM * MDIM;
  float* Rim   = ws + o; o += (size_t)MDIM * MDIM;
  float* augre = ws + o; o += (size_t)MDIM * 512;
  float* augim = ws + o; o += (size_t)MDIM * 512;
  float* RIre  = ws + o; o += (size_t)MDIM * MDIM;
  float* RIim  = ws + o; o += (size_t)MDIM * MDIM;
  float* Gre   = ws + o; o += (size_t)MDIM * MDIM;
  float* Gim   = ws + o; o += (size_t)MDIM * MDIM;
  float* ARre  = ws + o; o += (size_t)LDIM * MDIM;
  float* ARim  = ws + o; o += (size_t)LDIM * MDIM;
  float* Wre   = ws + o; o += (size_t)LDIM * MDIM;
  float* Wim   = ws + o; o += (size_t)LDIM * MDIM;
  float* qarr  = ws + o; o += LDIM;
  float* darr  = ws + o; o += LDIM;
  float* varr  = ws + o; o += LDIM;
  float* uarr  = ws + o; o += LDIM;
  float* qvarr = ws + o; o += LDIM;
  float* kvarr = ws + o; o += LDIM;
  float* vvarr = ws + o; o += LDIM;
  float* atarr = ws + o; o += LDIM;

  // 0) AP = A diag(p)
  scale_ap<<<(MDIM * LDIM) / 256, 256, 0, stream>>>(A_re, A_im, gamma, APre, APim);
  // 1) R = AP A^H + ridge I          [256x256], K=4096
  gemm_nt_conj<<<dim3(MDIM / 32, MDIM / 16), 32, 0, stream>>>(
      APre, APim, A_re, A_im, Rre, Rim, MDIM, LDIM, RIDGE);
  // 2) G = X X^H                     [256x256], K=2048
  gemm_nt_conj<<<dim3(MDIM / 32, MDIM / 16), 32, 0, stream>>>(
      X_re, X_im, X_re, X_im, Gre, Gim, MDIM, TDIM, 0.0f);
  // 3) Rinv
  invert_gj<<<1, 256, 0, stream>>>(Rre, Rim, augre, augim, RIre, RIim);
  // 4) ARi = A^H Rinv                [4096x256], K=256 (B panel via TDM->LDS)
  gemm_ah_b<<<dim3(MDIM / 16, LDIM / 128), 256, 0, stream>>>(
      A_re, A_im, RIre, RIim, ARre, ARim);
  // 5) W = ARi G                     [4096x256], K=256 (B panel via TDM->LDS)
  gemm_nn<<<dim3(MDIM / 16, LDIM / 128), 256, 0, stream>>>(
      ARre, ARim, Gre, Gim, Wre, Wim);
  // 6) q, d rowwise reductions
  qd_kernel<<<LDIM, 256, 0, stream>>>(Wre, Wim, ARre, ARim, A_re, A_im, qarr, darr);
  // 7) v, u
  uv_kernel<<<LDIM / 256, 256, 0, stream>>>(gamma, qarr, darr, delta, varr, uarr);
  // 8) LayerNorm + q/k/v projections
  ln_prep<<<1, 1024, 0, stream>>>(uarr, ln_w, ln_b, in_w, in_b, qvarr, kvarr, vvarr);
  // 9) attention (rank-1 scores, per-row softmax)
  attn_kernel<<<LDIM, 256, 0, stream>>>(qvarr, kvarr, vvarr, out_w, out_b, atarr);
  // 10) gate matvec + final elementwise
  gate_final<<<LDIM, 256, 0, stream>>>(gate_W, gate_b, uarr, varr, gamma, atarr,
                                       lmbda, out);
}